// QuantumLatent_65481071397455
// MI455X (gfx1250) — compile-verified
//
#include <hip/hip_runtime.h>
#include <stdint.h>

// ============================================================================
// 16-qubit batched statevector sim for MI455X (gfx1250, wave32).
//
// Circuit: RX(z+params) embed -> [RX(w_l) + CNOT-ring] x2 -> <Z_i>.
// Algebraic folding:
//   * embed RX + layer-0 RX fuse (RX(a)RX(b)=RX(a+b))  -> product state
//   * CNOT rings are GF(2)-linear index permutations:
//       ring 1 folded into the product-state build (index remap)
//       ring 2 folded into the <Z_i> sign masks (constexpr parity masks)
//   * only the layer-1 RX gates (16 of them) are simulated, LDS-tiled.
// Measurement: Walsh-Hadamard-style signed reduction done with chained
// V_WMMA_F32_16X16X4_F32 (exact f32), constant +-1 A-matrix.
// HBM traffic ~= 256 MB write + 256 MB read  ->  ~22 us roofline @ 23.3 TB/s.
// ============================================================================

#define NQ      16
#define DIM     65536     // 2^16 amplitudes per sample
#define TILE    4096      // 2^12 amplitudes per LDS tile
#define THREADS 256

typedef float v2f __attribute__((ext_vector_type(2)));
typedef float v8f __attribute__((ext_vector_type(8)));

// ---- compile-time circuit algebra (wire q <-> bit 15-q of the flat index) --
// One CNOT ring (gates C(q,q+1 mod 16) applied q=0..15) as an index map:
// psi_out[x] = psi_in[ringMap(x)].  Index transform of the LAST gate first.
constexpr uint32_t ringMap(uint32_t x) {
  uint32_t y = x & 0xFFFFu;
  y ^= (y & 1u) << 15;                      // C(15,0): ctrl bit0 -> tgt bit15
  for (int j = 1; j < 16; ++j)              // C(15-j,16-j): ctrl bit j -> bit j-1
    y ^= ((y >> j) & 1u) << (j - 1);
  return y & 0xFFFFu;
}
constexpr uint32_t ringInv(uint32_t y) {    // inverse (involutions, reversed)
  uint32_t x = y & 0xFFFFu;
  for (int j = 15; j >= 1; --j) x ^= ((x >> j) & 1u) << (j - 1);
  x ^= (x & 1u) << 15;
  return x & 0xFFFFu;
}
// <Z_i> after ring-2: sum_u p[u] * (-1)^popc(u & RM.m[i]); RM from linearity.
struct RMasks { uint32_t m[16]; };
constexpr RMasks makeR() {
  RMasks r{};
  for (int i = 0; i < 16; ++i) {
    const int jp = 15 - i;                  // wire i <-> bit 15-i
    uint32_t acc = 0;
    for (int k = 0; k < 16; ++k)
      acc |= ((ringInv(1u << k) >> jp) & 1u) << k;
    r.m[i] = acc;
  }
  return r;
}
constexpr RMasks RM = makeR();

__device__ __forceinline__ float psgn(uint32_t v) {
  return (__popc(v) & 1) ? -1.0f : 1.0f;
}

__global__ __launch_bounds__(THREADS)
void qsim_kernel(const float* __restrict__ z, const float* __restrict__ params,
                 const float* __restrict__ ent, float* __restrict__ out,
                 float2* __restrict__ ws, int sampleBase) {
  __shared__ float sRe[TILE];
  __shared__ float sIm[TILE];
  __shared__ float sC[16], sS[16];    // cos/sin of fused per-sample angles /2
  __shared__ float sGC[16], sGS[16];  // cos/sin of layer-1 entangler angles /2

  const int tid  = threadIdx.x;
  const int lane = tid & 31;
  const int b    = sampleBase + blockIdx.x;
  float2* st = ws + (size_t)blockIdx.x * DIM;   // 512 KB workspace slice

  if (tid < 16) {
    const int q = tid;
    const float th = z[b * NQ + q] + params[q] + ent[q];       // fused angles
    sC[q]  = cosf(th * 0.5f);
    sS[q]  = sinf(th * 0.5f);
    const float w1 = ent[NQ + q];                              // layer-1 RX
    sGC[q] = cosf(w1 * 0.5f);
    sGS[q] = sinf(w1 * 0.5f);
  }
  __syncthreads();

  // per-thread register copies of the product tables (constant-indexed)
  float cT[16], sT[16];
#pragma unroll
  for (int q = 0; q < 16; ++q) { cT[q] = sC[q]; sT[q] = sS[q]; }

  // ========== Phase A: build product state (ring-1 folded) + RX on bits 0..11
  for (int t4 = 0; t4 < 16; ++t4) {          // tile = x bits 12..15 fixed
    __syncthreads();
#pragma unroll
    for (int j = 0; j < 16; ++j) {
      const int lidx   = tid + THREADS * j;
      const uint32_t x = ((uint32_t)t4 << 12) | (uint32_t)lidx;
      const uint32_t y = ringMap(x);         // fold CNOT ring 1
      float mag = 1.0f;
#pragma unroll
      for (int q = 0; q < 16; ++q) {
        const uint32_t bit = (y >> (15 - q)) & 1u;
        mag *= bit ? sT[q] : cT[q];
      }
      const uint32_t k = (uint32_t)__popc(y) & 3u;   // phase (-i)^k
      float re = 0.0f, im = 0.0f;
      if (k == 0) re = mag; else if (k == 1) im = -mag;
      else if (k == 2) re = -mag; else im = mag;
      sRe[lidx] = re; sIm[lidx] = im;
    }
    // RX gates on local bits g=0..11  (wire 15-g), pairwise in LDS
    for (int g = 0; g < 12; ++g) {
      __syncthreads();
      const float cg = sGC[15 - g], sg = sGS[15 - g];
#pragma unroll
      for (int u = 0; u < 8; ++u) {
        const int idx = tid + THREADS * u;           // 2048 pairs
        const int p0  = ((idx >> g) << (g + 1)) | (idx & ((1 << g) - 1));
        const int p1  = p0 | (1 << g);
        const float a0r = sRe[p0], a0i = sIm[p0];
        const float a1r = sRe[p1], a1i = sIm[p1];
        sRe[p0] = cg * a0r + sg * a1i;
        sIm[p0] = cg * a0i - sg * a1r;
        sRe[p1] = cg * a1r + sg * a0i;
        sIm[p1] = cg * a1i - sg * a0r;
      }
    }
    __syncthreads();
#pragma unroll
    for (int j = 0; j < 16; ++j) {
      const int lidx = tid + THREADS * j;
      st[((uint32_t)t4 << 12) | (uint32_t)lidx] = make_float2(sRe[lidx], sIm[lidx]);
    }
  }
  __threadfence();
  __syncthreads();

  // ========== Phase C: retile, RX on bits 12..15, WMMA signed reduction =====
  // WMMA A operand: constant +-1 WHT rows.  16x4 f32 A layout:
  //   lanes 0-15: M=0-15, VGPR0 holds K=0 (K=2 in lanes 16-31), VGPR1 K=1/3.
  const int khx = (lane >= 16) ? 2 : 0;
  float ax[4], ay[4];
#pragma unroll
  for (int g = 0; g < 4; ++g) {
    ax[g] = psgn((uint32_t)((lane & 15) & (4 * g + khx)));
    ay[g] = psgn((uint32_t)((lane & 15) & (4 * g + khx + 1)));
  }
  // Fixup weights: pick WHT row v_i = (R_i>>4)&15 from the C-layout
  // (reg r=v_i&7, lane-half v_i>>3) and apply col sign over R_i low nibble.
  float W[16];
#pragma unroll
  for (int i = 0; i < 16; ++i) {
    const uint32_t vi = (RM.m[i] >> 4) & 15u;
    W[i] = ((uint32_t)(lane >> 4) == (vi >> 3))
               ? psgn((uint32_t)(lane & 15) & (RM.m[i] & 15u)) : 0.0f;
  }
  float ACC[16];
#pragma unroll
  for (int i = 0; i < 16; ++i) ACC[i] = 0.0f;

  const int wave = tid >> 5;
  for (int m = 0; m < 16; ++m) {             // m = x bits 8..11
    __syncthreads();
#pragma unroll
    for (int j = 0; j < 16; ++j) {
      const int lidx   = tid + THREADS * j;  // hi4 = lidx>>8, lo = lidx&255
      const uint32_t u = ((uint32_t)(lidx >> 8) << 12) |
                         ((uint32_t)m << 8) | (uint32_t)(lidx & 255);
      const float2 v = st[u];
      sRe[lidx] = v.x; sIm[lidx] = v.y;
      // Prefetch next m-tile into L2/WGP$ while we crunch this one
      // (global_prefetch_b8; fire-and-forget, no counters).
      if (m < 15) {
        const uint32_t un = u + 256u;        // same hi4, next m slice
        __builtin_prefetch((const void*)(st + un), 0, 3);
      }
    }
    // RX gates on local bits 8..11  (x bits 12..15, wires 3..0)
    for (int gl = 8; gl < 12; ++gl) {
      __syncthreads();
      const float cg = sGC[11 - gl], sg = sGS[11 - gl];
#pragma unroll
      for (int u = 0; u < 8; ++u) {
        const int idx = tid + THREADS * u;
        const int p0  = ((idx >> gl) << (gl + 1)) | (idx & ((1 << gl) - 1));
        const int p1  = p0 | (1 << gl);
        const float a0r = sRe[p0], a0i = sIm[p0];
        const float a1r = sRe[p1], a1i = sIm[p1];
        sRe[p0] = cg * a0r + sg * a1i;
        sIm[p0] = cg * a0i - sg * a1r;
        sRe[p1] = cg * a1r + sg * a0i;
        sIm[p1] = cg * a1i - sg * a0r;
      }
    }
    __syncthreads();
    // chunks of 256 probs: col = x bits 0..3, k = bits 4..7, chunk c = hi4
    for (int cc = 0; cc < 2; ++cc) {
      // c is wave-uniform; readfirstlane makes that visible to the compiler
      // so the per-chunk parity signs go through SALU instead of VALU.
      const int c    = __builtin_amdgcn_readfirstlane(wave + 8 * cc);
      const int base = c << 8;
      const int col  = lane & 15;
      v8f t = {};
#pragma unroll
      for (int g = 0; g < 4; ++g) {          // radix-16 WHT via 4 chained WMMAs
        const int kx = 4 * g + khx;
        v2f a; a.x = ax[g]; a.y = ay[g];
        v2f bb;
        { const int i0 = base + col + 16 * kx;
          const float r = sRe[i0], q = sIm[i0]; bb.x = r * r + q * q; }
        { const int i1 = base + col + 16 * (kx + 1);
          const float r = sRe[i1], q = sIm[i1]; bb.y = r * r + q * q; }
        t = __builtin_amdgcn_wmma_f32_16x16x4_f32(
                false, a, false, bb, (short)0, t, false, false);
      }
      const uint32_t yb = ((uint32_t)m << 8) | ((uint32_t)c << 12);
#pragma unroll
      for (int i = 0; i < 16; ++i) {
        const int par = __builtin_popcount(yb & RM.m[i]) & 1;   // uniform
        const float cs = par ? -1.0f : 1.0f;
        const int  r   = (int)((RM.m[i] >> 4) & 7u);   // compile-time row reg
        ACC[i] = fmaf(W[i] * cs, t[r], ACC[i]);
      }
    }
  }
  __syncthreads();
#pragma unroll
  for (int i = 0; i < 16; ++i) sRe[i * THREADS + tid] = ACC[i];
  __syncthreads();
  if (tid < 16) {
    float s = 0.0f;
    for (int j = 0; j < THREADS; ++j) s += sRe[tid * THREADS + j];
    out[b * NQ + tid] = s;
  }
}

extern "C" void kernel_launch(void* const* d_in, const int* in_sizes, int n_in,
                              void* d_out, int out_size, void* d_ws, size_t ws_size,
                              hipStream_t stream) {
  (void)n_in; (void)out_size;
  const float* z      = (const float*)d_in[0];
  const float* params = (const float*)d_in[1];
  const float* ent    = (const float*)d_in[2];
  float*  out = (float*)d_out;
  float2* ws  = (float2*)d_ws;

  const int    B   = in_sizes[0] / NQ;
  const size_t per = (size_t)DIM * sizeof(float2);      // 512 KB per sample
  int nper = (int)(ws_size / per);
  if (nper < 1) nper = 1;
  if (nper > B) nper = B;

  for (int base = 0; base < B; base += nper) {
    const int nb = (B - base < nper) ? (B - base) : nper;
    hipLaunchKernelGGL(qsim_kernel, dim3(nb), dim3(THREADS), 0, stream,
                       z, params, ent, out, ws, base);
  }
}